// MatchNet_4406636446104
// MI455X (gfx1250) — compile-verified
//
#include <hip/hip_runtime.h>

// ---------------------------------------------------------------------------
// MatchNet forward for MI455X (gfx1250, wave32).
// Pipeline: mutual_matching -> symmetric CenterPivot4D conv stack (VALU+LDS)
// -> mutual_matching -> masked softmax -> fp32 WMMA GEMM (v * attn^T).
// GEMM uses V_WMMA_F32_16X16X4_F32 fed from LDS tiles staged by the CDNA5
// Tensor Data Mover (tensor_load_to_lds + s_wait_tensorcnt).
// ---------------------------------------------------------------------------

typedef __attribute__((ext_vector_type(2))) float v2f;
typedef __attribute__((ext_vector_type(8))) float v8f;
typedef __attribute__((ext_vector_type(4))) unsigned int v4u;
typedef __attribute__((ext_vector_type(8))) int v8i;
typedef __attribute__((ext_vector_type(4))) int v4i;

#define EPSV 1e-5f
#define TEMPV 3.0f
#define NEG_BIG -3.402823466e38f

static constexpr int BATCH = 4;
static constexpr int QD = 1024;   // h*w
static constexpr int KD = 1024;   // hs*ws
static constexpr int CV = 256;
static constexpr int KSLAB = 64;  // K-slab staged in LDS per TDM transfer

// ---- Tensor Data Mover: 2-D tile (tile_d0 x tile_d1 elements, 4B each) ----
// Builds a D# descriptor per CDNA5 ISA ch.8 and issues TENSOR_LOAD_TO_LDS.
// amdgpu-toolchain (clang-23) 6-arg builtin form:
//   (v4u g0, v8i g1, v4i g2, v4i g3, v8i g4, i32 cpol)
__device__ __forceinline__ void tdm_load_2d(unsigned int lds_addr, const void* gptr,
                                            unsigned int tensor_d0, unsigned int tensor_d1,
                                            unsigned int tile_d0, unsigned int tile_d1,
                                            unsigned int stride0) {
    unsigned long long ga = (unsigned long long)(uintptr_t)gptr;
    v4u g0;
    g0.x = 1u;                                           // count=1 (valid user D#)
    g0.y = lds_addr;                                     // bits 63:32  lds_addr
    g0.z = (unsigned int)ga;                             // bits 95:64  global_addr lo
    g0.w = (unsigned int)((ga >> 32) & 0x1FFFFFFull)     // global_addr[56:32]
           | (2u << 30);                                 // type=2 ("image")
    v8i g1;
    g1[0] = (int)(2u << 16);                             // data_size=2 -> 4 bytes
    g1[1] = (int)(tensor_d0 << 16);                      // tensor_dim0[15:0]
    g1[2] = (int)((tensor_d0 >> 16) | (tensor_d1 << 16));// dim0 hi | dim1 lo
    g1[3] = (int)((tensor_d1 >> 16) | (tile_d0 << 16));  // dim1 hi | tile_dim0
    g1[4] = (int)(tile_d1 & 0xFFFFu);                    // tile_dim1, tile_dim2=0
    g1[5] = (int)stride0;                                // tensor_dim0_stride lo
    g1[6] = 0;                                           // stride0 hi | stride1 lo
    g1[7] = 0;
    v4i z4 = {0, 0, 0, 0};
    v8i z8 = {0, 0, 0, 0, 0, 0, 0, 0};
    __builtin_amdgcn_tensor_load_to_lds(g0, g1, z4, z4, z8, 0);
}

// ---- row max: amax[b*Q+q] = max_k x[b,q,k] --------------------------------
__global__ void k_rowmax(const float* __restrict__ x, float* __restrict__ amax) {
    int bq = blockIdx.x;                      // 0..4095
    const float* row = x + (size_t)bq * KD;
    float m = NEG_BIG;
    for (int k = threadIdx.x; k < KD; k += 256) m = fmaxf(m, row[k]);
    __shared__ float s[256];
    s[threadIdx.x] = m; __syncthreads();
    for (int off = 128; off > 0; off >>= 1) {
        if (threadIdx.x < off) s[threadIdx.x] = fmaxf(s[threadIdx.x], s[threadIdx.x + off]);
        __syncthreads();
    }
    if (threadIdx.x == 0) amax[bq] = s[0];
}

// ---- col max: bmax[b*K+k] = max_q x[b,q,k] --------------------------------
__global__ void k_colmax(const float* __restrict__ x, float* __restrict__ bmax) {
    int idx = blockIdx.x * 256 + threadIdx.x; // 0..4095  (b*1024 + k)
    int b = idx >> 10, k = idx & 1023;
    const float* base = x + (size_t)b * QD * KD + k;
    float m = NEG_BIG;
    for (int q = 0; q < QD; ++q) m = fmaxf(m, base[(size_t)q * KD]);
    bmax[idx] = m;
}

// ---- mutual matching apply: y = c * c/(amax+eps) * c/(bmax+eps) -----------
__global__ void k_mm_apply(const float* __restrict__ x, const float* __restrict__ amax,
                           const float* __restrict__ bmax, float* __restrict__ y, int total) {
    for (int i = blockIdx.x * blockDim.x + threadIdx.x; i < total; i += gridDim.x * blockDim.x) {
        int k = i & 1023, q = (i >> 10) & 1023, b = i >> 20;
        float c = x[i];
        y[i] = c * (c / (amax[b * 1024 + q] + EPSV)) * (c / (bmax[b * 1024 + k] + EPSV));
    }
}

// ---- tiled transpose: dst[b,k,q] = src[b,q,k] -----------------------------
__global__ void k_transpose(const float* __restrict__ src, float* __restrict__ dst) {
    __shared__ float tile[32][33];
    int b = blockIdx.z;
    int k0 = blockIdx.x * 32, q0 = blockIdx.y * 32;
    int tx = threadIdx.x, ty = threadIdx.y;
    tile[ty][tx] = src[((size_t)(b * QD + q0 + ty)) * KD + (k0 + tx)];
    __syncthreads();
    dst[((size_t)(b * KD + k0 + ty)) * QD + (q0 + tx)] = tile[tx][ty];
}

// ---- CenterPivotConv4d layer + bias + ReLU --------------------------------
// in:  [B, CI, 32,32, 32,32]   out: [B, CO, 32,32, 32,32]
// Block = (b, i, j); 1024 threads, thread t = support pixel (k,l).
template <int CI, int CO>
__global__ void __launch_bounds__(1024) k_cpconv(
        const float* __restrict__ in, float* __restrict__ out,
        const float* __restrict__ wq, const float* __restrict__ bq,
        const float* __restrict__ wsp, const float* __restrict__ bsp) {
    __shared__ float planes[9][1024];
    __shared__ float wq_s[CO * CI * 9];
    __shared__ float wsp_s[CO * CI * 9];
    __shared__ float bias_s[CO];

    int t = threadIdx.x;                 // support pixel index (k*32 + l)
    int j = blockIdx.x, i = blockIdx.y, b = blockIdx.z;

    for (int idx = t; idx < CO * CI * 9; idx += 1024) {
        wq_s[idx] = wq[idx];
        wsp_s[idx] = wsp[idx];
    }
    if (t < CO) bias_s[t] = bq[t] + bsp[t];

    float acc[CO];
#pragma unroll
    for (int co = 0; co < CO; ++co) acc[co] = 0.f;

    int kk = t >> 5, ll = t & 31;

    for (int ci = 0; ci < CI; ++ci) {
        __syncthreads();
#pragma unroll
        for (int p = 0; p < 9; ++p) {
            int dy = p / 3 - 1, dx = p % 3 - 1;
            int ii = i + dy, jj = j + dx;
            float v = 0.f;
            if (ii >= 0 && ii < 32 && jj >= 0 && jj < 32)
                v = in[(((size_t)(b * CI + ci) * 1024) + ii * 32 + jj) * 1024 + t];
            planes[p][t] = v;
        }
        __syncthreads();
        // conv over query dims: 9 neighbor planes at this support pixel
#pragma unroll
        for (int p = 0; p < 9; ++p) {
            float v = planes[p][t];
#pragma unroll
            for (int co = 0; co < CO; ++co)
                acc[co] += wq_s[(co * CI + ci) * 9 + p] * v;
        }
        // conv over support dims: shifted taps of the center plane
#pragma unroll
        for (int p = 0; p < 9; ++p) {
            int dy = p / 3 - 1, dx = p % 3 - 1;
            int k2 = kk + dy, l2 = ll + dx;
            float v = (k2 >= 0 && k2 < 32 && l2 >= 0 && l2 < 32) ? planes[4][k2 * 32 + l2] : 0.f;
#pragma unroll
            for (int co = 0; co < CO; ++co)
                acc[co] += wsp_s[(co * CI + ci) * 9 + p] * v;
        }
    }

    int ij = i * 32 + j;
#pragma unroll
    for (int co = 0; co < CO; ++co) {
        float r = acc[co] + bias_s[co];
        out[(((size_t)(b * CO + co) * 1024) + ij) * 1024 + t] = fmaxf(r, 0.f);
    }
}

// ---- out[b,q,k] = y1[b,q,k] + y2[b,k,q] -----------------------------------
__global__ void k_add_tr(const float* __restrict__ y1, const float* __restrict__ y2,
                         float* __restrict__ out, int total) {
    for (int i = blockIdx.x * blockDim.x + threadIdx.x; i < total; i += gridDim.x * blockDim.x) {
        int k = i & 1023, q = (i >> 10) & 1023, b = i >> 20;
        out[i] = y1[i] + y2[((size_t)(b * 1024 + k)) * 1024 + q];
    }
}

// ---- masked softmax over K (in place; row cached in registers) ------------
__global__ void k_softmax(float* __restrict__ logits, const int* __restrict__ mask) {
    int bq = blockIdx.x;
    int b = bq >> 10;
    float* row = logits + (size_t)bq * KD;
    const int* mrow = mask + b * KD;
    int t = threadIdx.x;

    float vals[4];
    float mx = NEG_BIG;
#pragma unroll
    for (int s = 0; s < 4; ++s) {
        int k = t + s * 256;
        float x = mrow[k] ? 1e-4f : row[k];
        x *= TEMPV;
        vals[s] = x;
        mx = fmaxf(mx, x);
    }
    __shared__ float red[256];
    red[t] = mx; __syncthreads();
    for (int off = 128; off > 0; off >>= 1) {
        if (t < off) red[t] = fmaxf(red[t], red[t + off]);
        __syncthreads();
    }
    mx = red[0]; __syncthreads();

    float sum = 0.f;
#pragma unroll
    for (int s = 0; s < 4; ++s) { vals[s] = __expf(vals[s] - mx); sum += vals[s]; }
    red[t] = sum; __syncthreads();
    for (int off = 128; off > 0; off >>= 1) {
        if (t < off) red[t] += red[t + off];
        __syncthreads();
    }
    float inv = 1.f / red[0];
#pragma unroll
    for (int s = 0; s < 4; ++s) row[t + s * 256] = vals[s] * inv;
}

// ---- fp32 WMMA GEMM with TDM staging --------------------------------------
// out[b,m,q] = sum_k v[b,m,k] * attn[b,q,k].
// Block = (ntGroup, mt, b): 8 waves, wave w computes the 16x16 tile
// (m0..m0+15) x (q0..q0+15), q0 = (ntGroup*8+w)*16. All waves share the A
// (V) tile; wave 0 DMAs it to LDS via the Tensor Data Mover, each wave DMAs
// its own attn tile slab. WMMA fragments are then read from LDS.
// Fragment layout per CDNA5 ISA 7.12.2:
//   A (16x4): lanes 0-15 -> {K=0,K=1}, lanes 16-31 -> {K=2,K=3}, M=lane%16
//   B (4x16): vgpr0 -> lanes0-15 K=0 / lanes16-31 K=1; vgpr1 -> K=2 / K=3
//   C/D: vgpr r -> M = r (lanes 0-15) or r+8 (lanes 16-31), N = lane%16
__global__ void __launch_bounds__(256) k_wmma_gemm(const float* __restrict__ v,
                                                   const float* __restrict__ attn,
                                                   float* __restrict__ out) {
    __shared__ float As[16][KSLAB];
    __shared__ float Bs[8][16][KSLAB];

    int lane = threadIdx.x & 31;
    int w    = threadIdx.x >> 5;      // wave in block: 0..7
    int ng   = blockIdx.x;            // 0..7
    int mt   = blockIdx.y;            // 0..15
    int b    = blockIdx.z;            // 0..3

    int m0 = mt * 16;
    int q0 = (ng * 8 + w) * 16;
    int half = lane >> 4;
    int l16  = lane & 15;

    const float* Vb = v + ((size_t)b * CV + m0) * KD;     // 16 rows of V
    const float* Ab = attn + ((size_t)b * QD + q0) * KD;  // 16 rows of attn

    unsigned int ldsA = (unsigned int)(uintptr_t)&As[0][0];
    unsigned int ldsB = (unsigned int)(uintptr_t)&Bs[w][0][0];

    v8f acc = {};
    for (int k0 = 0; k0 < KD; k0 += KSLAB) {
        __syncthreads();  // previous slab fully consumed
        tdm_load_2d(ldsB, Ab + k0, KD, QD, KSLAB, 16, KD);             // attn tile
        if (w == 0) tdm_load_2d(ldsA, Vb + k0, KD, CV, KSLAB, 16, KD); // shared V tile
        __builtin_amdgcn_s_wait_tensorcnt(0);
        __syncthreads();  // A slab visible to all waves

#pragma unroll
        for (int k = 0; k < KSLAB; k += 4) {
            v2f a, bb;
            a.x  = As[l16][k + half * 2];
            a.y  = As[l16][k + half * 2 + 1];
            bb.x = Bs[w][l16][k + half];
            bb.y = Bs[w][l16][k + half + 2];
            acc = __builtin_amdgcn_wmma_f32_16x16x4_f32(
                /*neg_a=*/false, a, /*neg_b=*/false, bb,
                /*c_mod=*/(short)0, acc, /*reuse_a=*/false, /*reuse_b=*/false);
        }
    }

#pragma unroll
    for (int r = 0; r < 8; ++r) {
        int m = m0 + r + half * 8;
        out[((size_t)b * CV + m) * QD + q0 + l16] = acc[r];
    }
}

// ---------------------------------------------------------------------------
extern "C" void kernel_launch(void* const* d_in, const int* in_sizes, int n_in,
                              void* d_out, int out_size, void* d_ws, size_t ws_size,
                              hipStream_t stream) {
    const float* corr = (const float*)d_in[0];
    const float* vmat = (const float*)d_in[1];
    const int*   mask = (const int*)d_in[2];
    const float* wq0 = (const float*)d_in[3];  const float* bq0 = (const float*)d_in[4];
    const float* ws0 = (const float*)d_in[5];  const float* bs0 = (const float*)d_in[6];
    const float* wq1 = (const float*)d_in[7];  const float* bq1 = (const float*)d_in[8];
    const float* ws1 = (const float*)d_in[9];  const float* bs1 = (const float*)d_in[10];
    const float* wq2 = (const float*)d_in[11]; const float* bq2 = (const float*)d_in[12];
    const float* ws2 = (const float*)d_in[13]; const float* bs2 = (const float*)d_in[14];

    const size_t N1  = (size_t)BATCH * QD * KD;        // 4,194,304 floats
    const size_t N10 = (size_t)BATCH * 10 * QD * KD;   // 41,943,040 floats

    float* ws = (float*)d_ws;
    float* t1    = ws;              // mm1 output / reused as y2
    float* tT    = t1 + N1;         // transposed copy
    float* bufA  = tT + N1;         // 10-channel intermediate A
    float* bufB  = bufA + N10;      // 10-channel intermediate B
    float* y1    = bufB + N10;      // stack-1 output / reused as logits
    float* ysum  = y1 + N1;         // symmetric sum
    float* amax  = ysum + N1;       // [B*1024]
    float* bmax  = amax + BATCH * 1024;

    const int total = (int)N1;
    dim3 convGrid(32, 32, BATCH);
    dim3 trGrid(32, 32, BATCH), trBlock(32, 32, 1);

    // mutual matching #1
    k_rowmax<<<BATCH * QD, 256, 0, stream>>>(corr, amax);
    k_colmax<<<BATCH * KD / 256, 256, 0, stream>>>(corr, bmax);
    k_mm_apply<<<2048, 256, 0, stream>>>(corr, amax, bmax, t1, total);

    // symmetric conv stacks
    k_transpose<<<trGrid, trBlock, 0, stream>>>(t1, tT);
    k_cpconv<1, 10><<<convGrid, 1024, 0, stream>>>(t1, bufA, wq0, bq0, ws0, bs0);
    k_cpconv<10, 10><<<convGrid, 1024, 0, stream>>>(bufA, bufB, wq1, bq1, ws1, bs1);
    k_cpconv<10, 1><<<convGrid, 1024, 0, stream>>>(bufB, y1, wq2, bq2, ws2, bs2);
    k_cpconv<1, 10><<<convGrid, 1024, 0, stream>>>(tT, bufA, wq0, bq0, ws0, bs0);
    k_cpconv<10, 10><<<convGrid, 1024, 0, stream>>>(bufA, bufB, wq1, bq1, ws1, bs1);
    k_cpconv<10, 1><<<convGrid, 1024, 0, stream>>>(bufB, t1 /*y2*/, wq2, bq2, ws2, bs2);
    k_add_tr<<<2048, 256, 0, stream>>>(y1, t1, ysum, total);

    // mutual matching #2
    k_rowmax<<<BATCH * QD, 256, 0, stream>>>(ysum, amax);
    k_colmax<<<BATCH * KD / 256, 256, 0, stream>>>(ysum, bmax);
    k_mm_apply<<<2048, 256, 0, stream>>>(ysum, amax, bmax, y1 /*logits*/, total);

    // masked softmax (in place), then TDM-staged fp32 WMMA attention GEMM
    k_softmax<<<BATCH * QD, 256, 0, stream>>>(y1, mask);
    k_wmma_gemm<<<dim3(8, 16, BATCH), 256, 0, stream>>>(vmat, y1, (float*)d_out);

    (void)in_sizes; (void)n_in; (void)out_size; (void)ws_size;
}